// HEALPixPadding_90975997264310
// MI455X (gfx1250) — compile-verified
//
#include <hip/hip_runtime.h>
#include <hip/hip_bf16.h>
#include <stdint.h>
#include <stddef.h>

// ---------------- problem constants (match reference) ----------------
#define PW 2                       // padding width
#define NS 128                     // face edge (nside)
#define NC 32                      // channels
#define HP (NS + 2 * PW)           // 132 padded edge
#define FACE_IN  (NS * NS)         // 16384 floats per (bf,ch) input image
#define FACE_OUT (HP * HP)         // 17424 floats per (bf,ch) output image
#define FSTRIDE  ((size_t)NC * FACE_IN)   // face-to-face stride (fixed b,ch), floats
#define HALO_N   (4 * HP + 4 * NS)        // 1040 halo elements per image

// ---------------- CDNA5 async global<->LDS availability ----------------
#if defined(__AMDGCN__) && __has_builtin(__builtin_amdgcn_global_load_async_to_lds_b64) && \
    __has_builtin(__builtin_amdgcn_global_store_async_from_lds_b64)
#define HAVE_ASYNC 1
#else
#define HAVE_ASYNC 0
#endif

#if HAVE_ASYNC
typedef int v2i __attribute__((ext_vector_type(2)));
typedef __attribute__((address_space(1))) v2i* gptr_v2i;   // global (AS1) pointer to 64-bit chunk
typedef __attribute__((address_space(3))) v2i* lptr_v2i;   // LDS (AS3) pointer to 64-bit chunk
#endif

// ---------------- neighbor tables ----------------
// kNb[fo] = { t, tl, lf, bl, b, br, rg, tr }; -1 => interpolated corner (pe faces)
__device__ __constant__ int kType[12] = {0, 0, 0, 0, 1, 1, 1, 1, 2, 2, 2, 2};
__device__ __constant__ int kNb[12][8] = {
    {1, 2, 3, 3, 4, 8, 5, 1},      // f0  (pn)
    {2, 3, 0, 0, 5, 9, 6, 2},      // f1
    {3, 0, 1, 1, 6, 10, 7, 3},     // f2
    {0, 1, 2, 2, 7, 11, 4, 0},     // f3
    {0, -1, 3, 7, 11, -1, 8, 5},   // f4  (pe; tl/br interpolated from t/lf, b/rg)
    {1, -1, 0, 4, 8, -1, 9, 6},    // f5
    {2, -1, 1, 5, 9, -1, 10, 7},   // f6
    {3, -1, 2, 6, 10, -1, 11, 4},  // f7
    {5, 0, 4, 11, 11, 10, 9, 9},   // f8  (ps)
    {6, 1, 5, 8, 8, 11, 10, 10},   // f9
    {7, 2, 6, 9, 9, 8, 11, 11},    // f10
    {4, 3, 7, 10, 10, 9, 8, 8},    // f11
};

__device__ __forceinline__ float face_at(const float* xb, int g, int r, int c) {
    return xb[(size_t)g * FSTRIDE + (size_t)r * NS + (size_t)c];
}

// Value of padded image (face fo) at padded coords (R, Cc), R,Cc in [0, HP)
__device__ float halo_value(const float* xb, int fo, int R, int Cc) {
    const int type = kType[fo];
    const int t  = kNb[fo][0], tl = kNb[fo][1], lf = kNb[fo][2], bl = kNb[fo][3];
    const int bb = kNb[fo][4], br = kNb[fo][5], rg = kNb[fo][6], tr = kNb[fo][7];
    const int r = R - PW, c = Cc - PW;
    const bool top = (r < 0), bot = (r >= NS), lft = (c < 0), rgt = (c >= NS);

    if (!top && !bot) {
        if (lft) {  // left strip, j = Cc in [0,PW)
            const int j = Cc;
            return (type == 0) ? face_at(xb, lf, PW - 1 - j, r)        // pn: rot(lft,-1)[:, -p:]
                               : face_at(xb, lf, r, NS - PW + j);      // pe/ps: lft[:, -p:]
        }
        // right strip, j = c-NS in [0,PW)
        const int j = c - NS;
        return (type == 2) ? face_at(xb, rg, NS - 1 - j, r)            // ps: rot(rgt,-1)[:, :p]
                           : face_at(xb, rg, r, j);                    // pn/pe: rgt[:, :p]
    }
    if (top) {
        const int j = R;  // strip row in [0,PW)
        if (!lft && !rgt) {
            return (type == 0) ? face_at(xb, t, c, PW - 1 - j)         // pn: rot(t,1)[-p:, :]
                               : face_at(xb, t, NS - PW + j, c);       // pe/ps: t[-p:, :]
        }
        if (rgt) {  // top-right corner: tr[-p:, :p] for all types
            const int i2 = c - NS;
            return face_at(xb, tr, NS - PW + j, i2);
        }
        // top-left corner, block coords (j, i2)
        const int i2 = Cc;
        if (type == 0) return face_at(xb, tl, PW - 1 - j, PW - 1 - i2);    // pn: rot(tl,2)[-p:, -p:]
        if (type == 2) return face_at(xb, tl, NS - PW + j, NS - PW + i2);  // ps: tl[-p:, -p:]
        // pe: interpolated _tl(t, lf)
        if (j == i2)
            return 0.5f * face_at(xb, t, NS - PW + j, 0) + 0.5f * face_at(xb, lf, 0, NS - PW + j);
        if (i2 > j) return face_at(xb, t, NS - PW + j, i2 - j - 1);
        return face_at(xb, lf, j - i2 - 1, NS - PW + i2);
    }
    // bottom
    const int i = r - NS;  // strip row in [0,PW)
    if (!lft && !rgt) {
        return (type == 2) ? face_at(xb, bb, c, NS - 1 - i)            // ps: rot(b,1)[:p, :]
                           : face_at(xb, bb, i, c);                    // pn/pe: b[:p, :]
    }
    if (lft) {  // bottom-left corner: bl[:p, -p:] for all types
        const int j2 = Cc;
        return face_at(xb, bl, i, NS - PW + j2);
    }
    // bottom-right corner, block coords (i, j2)
    const int j2 = c - NS;
    if (type == 0) return face_at(xb, br, i, j2);                      // pn: br[:p, :p]
    if (type == 2) return face_at(xb, br, NS - 1 - i, NS - 1 - j2);    // ps: rot(br,2)[:p, :p]
    // pe: interpolated _br(bb, rg)
    if (i == j2)
        return 0.5f * face_at(xb, bb, i, NS - 1) + 0.5f * face_at(xb, rg, NS - 1, j2);
    if (j2 > i) return face_at(xb, rg, NS - j2 + i, j2);
    return face_at(xb, bb, i, NS - i + j2);
}

// ---------------- interior: pure shifted copy, 94% of traffic ----------------
// block = 256 threads, grid = (NS/32, nImg). Each block moves 32 rows of one image.
// Output interior starts at +2 floats -> 8-byte aligned, so move B64 chunks.
__global__ void healpix_interior(const float* __restrict__ x, float* __restrict__ out) {
    const int img = blockIdx.y;                       // (b*12+fo)*NC + ch
    const int r0  = blockIdx.x * 32;
    const size_t inBase  = (size_t)img * FACE_IN  + (size_t)r0 * NS;
    const size_t outBase = (size_t)img * FACE_OUT + (size_t)(r0 + PW) * HP + PW;
    const int tid = threadIdx.x;
#if HAVE_ASYNC
    __shared__ float lds[32 * NS];                    // 16 KB staging
    #pragma unroll
    for (int k = 0; k < 8; ++k) {
        const int e  = k * 256 + tid;                 // b64 chunk id in [0, 2048)
        const int rr = e >> 6, cc = e & 63;
        const float* gsrc = x + inBase + (size_t)rr * NS + (size_t)(cc * 2);
        __builtin_amdgcn_global_load_async_to_lds_b64(
            (gptr_v2i)gsrc, (lptr_v2i)&lds[e * 2], 0 /*offset*/, 0 /*cpol*/);
    }
#if __has_builtin(__builtin_amdgcn_s_wait_asynccnt)
    __builtin_amdgcn_s_wait_asynccnt(0);
#else
    asm volatile("s_wait_asynccnt 0" ::: "memory");
#endif
    #pragma unroll
    for (int k = 0; k < 8; ++k) {
        const int e  = k * 256 + tid;
        const int rr = e >> 6, cc = e & 63;
        float* gdst = out + outBase + (size_t)rr * HP + (size_t)(cc * 2);
        __builtin_amdgcn_global_store_async_from_lds_b64(
            (gptr_v2i)gdst, (lptr_v2i)&lds[e * 2], 0 /*offset*/, 0 /*cpol*/);
    }
#else
    #pragma unroll
    for (int k = 0; k < 8; ++k) {
        const int e  = k * 256 + tid;
        const int rr = e >> 6, cc = e & 63;
        const float2 v = *(const float2*)(x + inBase + (size_t)rr * NS + (size_t)(cc * 2));
        *(float2*)(out + outBase + (size_t)rr * HP + (size_t)(cc * 2)) = v;
    }
#endif
}

// ---------------- halo: 1040 gathered elements per image, 6% of traffic ----------------
__global__ void healpix_halo(const float* __restrict__ x, float* __restrict__ out) {
    const int img = blockIdx.y;                       // (b*12+fo)*NC + ch
    const int h = blockIdx.x * blockDim.x + threadIdx.x;
    if (h >= HALO_N) return;
    const int ch = img % NC;
    const int bf = img / NC;
    const int fo = bf % 12;
    const int b  = bf / 12;
    int R, Cc;
    if (h < 2 * HP) {                                 // top 2 full rows
        R = h / HP; Cc = h % HP;
    } else if (h < 4 * HP) {                          // bottom 2 full rows
        const int u = h - 2 * HP; R = (NS + PW) + u / HP; Cc = u % HP;
    } else if (h < 4 * HP + 2 * NS) {                 // left 2 cols (middle rows)
        const int u = h - 4 * HP; R = PW + (u >> 1); Cc = u & 1;
    } else {                                          // right 2 cols (middle rows)
        const int u = h - 4 * HP - 2 * NS; R = PW + (u >> 1); Cc = NS + PW + (u & 1);
    }
    const float* xb = x + (size_t)b * 12 * FSTRIDE + (size_t)ch * FACE_IN;
    out[(size_t)img * FACE_OUT + (size_t)R * HP + (size_t)Cc] = halo_value(xb, fo, R, Cc);
}

extern "C" void kernel_launch(void* const* d_in, const int* in_sizes, int n_in,
                              void* d_out, int out_size, void* d_ws, size_t ws_size,
                              hipStream_t stream) {
    const float* x = (const float*)d_in[0];
    float* out = (float*)d_out;
    const int nImg = in_sizes[0] / FACE_IN;           // B*12*NC  (= 3072 for B=8)

    dim3 gi(NS / 32, nImg);
    healpix_interior<<<gi, 256, 0, stream>>>(x, out);

    dim3 gh((HALO_N + 255) / 256, nImg);
    healpix_halo<<<gh, 256, 0, stream>>>(x, out);
}